// MeanAggregatorHead_8065948582554
// MI455X (gfx1250) — compile-verified
//
#include <hip/hip_runtime.h>

// MeanAggregator: out[b,:] = mean over K=32 gathered 512B rows of a [N,128] fp32 table.
// Pure gather-reduce, AI ~0.25 FLOP/byte -> HBM-bound (floor ~73us @ 23.3 TB/s).
// Strategy: CDNA5 async global->LDS copies (GLOBAL_LOAD_ASYNC_TO_LDS_B128, GVS form,
// no VGPR destinations), software-pipelined 4x8 groups over a 16-slot LDS ring per
// wave, scalar (SALU) row addressing via v_readlane.

constexpr int kK = 32;             // neighbors per node
constexpr int kD = 128;            // feature dim
constexpr int kWavesPerBlock = 8;  // 256 threads, wave32
constexpr int kSlots = 16;         // LDS row slots per wave (ring, 2 groups of 8)
constexpr int kSlotBytes = kD * 4; // 512 B per embedding row

// One async issue: gathered row -> this wave's LDS slot SLOT.
// GVS addressing: global = SADDR(row base, scalar) + VADDR(lane*16).
// LDS dest VGPR = laddr_base(+escaped shared ptr) + SLOT*512: passing the
// pointer-derived offset into the asm makes smem address-taken, so the
// compiler must assume the asm writes it (keeps the DS consume loads alive).
template <int SLOT>
__device__ __forceinline__ void issue_async(const float* table, int row,
                                            unsigned laddr_base, unsigned voff) {
    const char* sbase = (const char*)table + (((size_t)(unsigned)row) << 9); // row*512B
    const unsigned laddr = laddr_base + (unsigned)(SLOT * kSlotBytes);
    asm volatile("global_load_async_to_lds_b128 %0, %1, %2"
                 :: "v"(laddr), "v"(voff), "s"(sbase)
                 : "memory");
}

// Issue 8 neighbors KB..KB+7 into slots SB..SB+7 (all immediates compile-time).
#define ISSUE8(KB, SB)                                                                        \
    issue_async<(SB) + 0>(table, __builtin_amdgcn_readlane(myidx, (KB) + 0), laddr_base, voff); \
    issue_async<(SB) + 1>(table, __builtin_amdgcn_readlane(myidx, (KB) + 1), laddr_base, voff); \
    issue_async<(SB) + 2>(table, __builtin_amdgcn_readlane(myidx, (KB) + 2), laddr_base, voff); \
    issue_async<(SB) + 3>(table, __builtin_amdgcn_readlane(myidx, (KB) + 3), laddr_base, voff); \
    issue_async<(SB) + 4>(table, __builtin_amdgcn_readlane(myidx, (KB) + 4), laddr_base, voff); \
    issue_async<(SB) + 5>(table, __builtin_amdgcn_readlane(myidx, (KB) + 5), laddr_base, voff); \
    issue_async<(SB) + 6>(table, __builtin_amdgcn_readlane(myidx, (KB) + 6), laddr_base, voff); \
    issue_async<(SB) + 7>(table, __builtin_amdgcn_readlane(myidx, (KB) + 7), laddr_base, voff);

// Accumulate 8 staged rows (slots SB..SB+7): ds_load_b128 + v_add_f32 per lane chunk.
#define CONSUME8(SB)                                                            \
    _Pragma("unroll")                                                           \
    for (int s_ = 0; s_ < 8; ++s_) {                                            \
        const float4 v_ = sv[((SB) + s_) * (kD / 4) + lane];                    \
        acc.x += v_.x; acc.y += v_.y; acc.z += v_.z; acc.w += v_.w;             \
    }

__global__ __launch_bounds__(256)
void mean_agg_async_kernel(const float* __restrict__ table,
                           const int* __restrict__ idx,
                           float* __restrict__ out,
                           int B)
{
    // 8 waves * 16 slots * 512B = 64 KB static LDS (WGP pool is 320 KB -> ~5 WGs).
    __shared__ __align__(16) float smem[kWavesPerBlock * kSlots * kD];

    const int lane = threadIdx.x & 31;
    const int wid  = threadIdx.x >> 5;
    const int b    = blockIdx.x * kWavesPerBlock + wid;   // wave-uniform
    if (b >= B) return;

    // Lane l holds neighbor l's row index; broadcast via v_readlane (-> SGPR).
    const int myidx = idx[b * kK + lane];

    // Low 32 bits of a generic LDS address == LDS byte offset (aperture layout).
    // The ptrtoint escape is what keeps smem's loads from folding to undef.
    float* const wave_lds = &smem[wid * kSlots * kD];
    const unsigned laddr_base = (unsigned)(uintptr_t)wave_lds + (unsigned)(lane * 16);
    const unsigned voff       = (unsigned)(lane * 16);    // 32 lanes x 16B = one 512B row
    const float4* sv = (const float4*)wave_lds;

    float4 acc = make_float4(0.f, 0.f, 0.f, 0.f);

    // ---- software pipeline: 4 groups of 8, ring of 16 slots, 8-16 rows in flight ----
    ISSUE8(0, 0);                                // g0 -> slots 0..7
    ISSUE8(8, 8);                                // g1 -> slots 8..15   (16 outstanding)

    asm volatile("s_wait_asynccnt 0x8" ::: "memory");   // >=8 done (in-order) => g0 landed
    CONSUME8(0);
    asm volatile("s_wait_dscnt 0x0" ::: "memory");      // our DS reads retired before reuse
    ISSUE8(16, 0);                               // g2 -> slots 0..7

    asm volatile("s_wait_asynccnt 0x8" ::: "memory");   // issued 24, <=8 => g1 landed
    CONSUME8(8);
    asm volatile("s_wait_dscnt 0x0" ::: "memory");
    ISSUE8(24, 8);                               // g3 -> slots 8..15

    asm volatile("s_wait_asynccnt 0x8" ::: "memory");   // issued 32, <=8 => g2 landed
    CONSUME8(0);

    asm volatile("s_wait_asynccnt 0x0" ::: "memory");   // g3 landed
    CONSUME8(8);

    // Mean, then coalesced 16B/lane store (wave writes the full 512B output row).
    const float scale = 1.0f / (float)kK;
    float4 r;
    r.x = acc.x * scale; r.y = acc.y * scale;
    r.z = acc.z * scale; r.w = acc.w * scale;
    ((float4*)(out + (size_t)b * kD))[lane] = r;
}

extern "C" void kernel_launch(void* const* d_in, const int* in_sizes, int n_in,
                              void* d_out, int out_size, void* d_ws, size_t ws_size,
                              hipStream_t stream) {
    (void)in_sizes; (void)n_in; (void)d_ws; (void)ws_size;
    const float* table = (const float*)d_in[0];   // embed_table [N,128] fp32
    const int*   idx   = (const int*)d_in[1];     // neigh_idx   [B,32]
    float*       out   = (float*)d_out;           // [B,128] fp32

    const int B = out_size / kD;                  // deterministic
    const int blocks = (B + kWavesPerBlock - 1) / kWavesPerBlock;

    hipLaunchKernelGGL(mean_agg_async_kernel,
                       dim3(blocks), dim3(kWavesPerBlock * 32), 0, stream,
                       table, idx, out, B);
}